// HierarchicalLinearMemoryManager_87926570483969
// MI455X (gfx1250) — compile-verified
//
#include <hip/hip_runtime.h>
#include <hip/hip_bf16.h>
#include <stdint.h>

// Problem constants (B, R, C, RES, M from the reference)
#define Bn   16
#define Rn   16
#define Cn   640
#define RESn 16
#define Mn   32

#define SLOT_F   (Cn * RESn * RESn)          // 163840 floats per value slot
#define SLOT_V4  (SLOT_F / 4)                // 40960 float4 per slot
#define VAL_CHUNKS 40                        // chunks per slot (grid.y)
#define V4_PER_BLOCK (SLOT_V4 / VAL_CHUNKS)  // 1024 float4 per block
#define VTHREADS 256
#define VITERS   (V4_PER_BLOCK / VTHREADS)   // 4

#define KEYS_F ((size_t)Bn * Mn * Cn)        // 327680
#define VALS_F ((size_t)Bn * Mn * SLOT_F)    // 83886080
#define OUT_KEYS_OFF  ((size_t)0)
#define OUT_VALS_OFF  (KEYS_F)
#define OUT_VALID_OFF (KEYS_F + VALS_F)
#define OUT_COUNT_OFF (OUT_VALID_OFF + (size_t)Bn * Mn)

// CDNA5 async global<->LDS DMA path (ASYNCcnt-tracked, no VGPR data movement)
#if defined(__HIP_DEVICE_COMPILE__) && defined(__gfx1250__) &&                 \
    __has_builtin(__builtin_amdgcn_global_load_async_to_lds_b128) &&           \
    __has_builtin(__builtin_amdgcn_global_store_async_from_lds_b128)
#define USE_ASYNC_LDS 1
#else
#define USE_ASYNC_LDS 0
#endif

// Builtin signature (confirmed by hipcc diagnostics):
//   arg0 = addrspace(1) int4* (global), arg1 = addrspace(3) int4* (LDS),
//   arg2 = imm offset, arg3 = imm cpol.
typedef int v4i __attribute__((vector_size(16)));
typedef __attribute__((address_space(1))) v4i* gptr_v4i;
typedef __attribute__((address_space(3))) v4i* lptr_v4i;

__device__ __forceinline__ void wait_async0() {
#if defined(__HIP_DEVICE_COMPILE__) && defined(__gfx1250__)
#if __has_builtin(__builtin_amdgcn_s_wait_asynccnt)
  __builtin_amdgcn_s_wait_asynccnt(0);
#else
  asm volatile("s_wait_asynccnt 0" ::: "memory");
#endif
#endif
}

// Per-slot ring-buffer metadata: is slot m of batch b hit by the scatter, and
// if so by which source ref j; also first-valid index and num_valid.
__device__ __forceinline__ void slot_meta(const int* __restrict__ src_valid,
                                          const int* __restrict__ mem_count,
                                          int b, int m,
                                          int& hit, int& j, int& first, int& nv) {
  const int cnt = mem_count[b];
  const int r = ((m - cnt) % Mn + Mn) % Mn;   // rank that would map to slot m
  j = -1;
  first = -1;
  int rank = 0;
  #pragma unroll
  for (int i = 0; i < Rn; ++i) {
    if (src_valid[b * Rn + i] != 0) {
      if (first < 0) first = i;
      if (rank == r) j = i;
      ++rank;
    }
  }
  nv = rank;
  hit = (r < nv) ? 1 : 0;
  if (first < 0) first = 0;  // argmax of all-false mask is 0 (never dereferenced for a hit)
}

// ---- keys + valid + count: 512 blocks x 160 threads (one float4/thread) ----
__global__ __launch_bounds__(Cn / 4) void hlmm_keys_kernel(
    const float* __restrict__ src_keys, const float* __restrict__ mem_keys,
    const int* __restrict__ src_valid, const int* __restrict__ mem_valid,
    const int* __restrict__ mem_count, float* __restrict__ out) {
  const int slot = blockIdx.x;           // b*Mn + m
  const int b = slot / Mn, m = slot % Mn;
  int hit, j, first, nv;
  slot_meta(src_valid, mem_count, b, m, hit, j, first, nv);

  const float4* s4 =
      hit ? (const float4*)(src_keys + ((size_t)b * Rn + (size_t)j) * Cn)
          : (const float4*)(mem_keys + (size_t)slot * Cn);
  float4* d4 = (float4*)(out + OUT_KEYS_OFF + (size_t)slot * Cn);
  d4[threadIdx.x] = s4[threadIdx.x];

  if (threadIdx.x == 0) {
    out[OUT_VALID_OFF + slot] = (hit || mem_valid[slot] != 0) ? 1.0f : 0.0f;
    if (m == 0) out[OUT_COUNT_OFF + b] = (float)(mem_count[b] + nv);
  }
}

// ---- values: 512 slots x 40 chunks, 256 threads; async LDS-staged b128 copy ----
__global__ __launch_bounds__(VTHREADS) void hlmm_values_kernel(
    const float* __restrict__ src_vals, const int* __restrict__ src_valid,
    const float* __restrict__ mem_vals, const int* __restrict__ mem_count,
    float* __restrict__ out) {
  const int slot = blockIdx.x;           // b*Mn + m
  const int b = slot / Mn, m = slot % Mn;
  int hit, j, first, nv;
  slot_meta(src_valid, mem_count, b, m, hit, j, first, nv);

  // Scatter hits always receive the FIRST valid ref's value tensor (reference quirk).
  const float4* src4 =
      hit ? (const float4*)(src_vals + ((size_t)b * Rn + (size_t)first) * SLOT_F)
          : (const float4*)(mem_vals + (size_t)slot * SLOT_F);
  float4* dst4 = (float4*)(out + OUT_VALS_OFF + (size_t)slot * SLOT_F);

  const int base = blockIdx.y * V4_PER_BLOCK + (int)threadIdx.x;

#if USE_ASYNC_LDS
  __shared__ float4 stage[VITERS][VTHREADS];  // 16 KB staging
  #pragma unroll
  for (int it = 0; it < VITERS; ++it) {
    // flat and global addresses coincide numerically on AMDGPU, so the
    // uintptr_t round-trip is an exact generic->AS1 conversion here.
    __builtin_amdgcn_global_load_async_to_lds_b128(
        (gptr_v4i)(uintptr_t)(src4 + base + it * VTHREADS),
        (lptr_v4i)&stage[it][threadIdx.x], 0, 0);
  }
  wait_async0();  // LDS data landed
  #pragma unroll
  for (int it = 0; it < VITERS; ++it) {
    __builtin_amdgcn_global_store_async_from_lds_b128(
        (gptr_v4i)(uintptr_t)(dst4 + base + it * VTHREADS),
        (lptr_v4i)&stage[it][threadIdx.x], 0, 0);
  }
  wait_async0();  // stores drained before wave exit
#else
  #pragma unroll
  for (int it = 0; it < VITERS; ++it) {
    dst4[base + it * VTHREADS] = src4[base + it * VTHREADS];
  }
#endif
}

extern "C" void kernel_launch(void* const* d_in, const int* in_sizes, int n_in,
                              void* d_out, int out_size, void* d_ws, size_t ws_size,
                              hipStream_t stream) {
  (void)in_sizes; (void)n_in; (void)out_size; (void)d_ws; (void)ws_size;
  const float* src_keys = (const float*)d_in[0];  // [B,R,C]
  const float* src_vals = (const float*)d_in[1];  // [B,R,C,RES,RES]
  const int*   src_valid = (const int*)d_in[2];   // [B,R] bool->int
  const float* mem_keys = (const float*)d_in[3];  // [B,M,C]
  const float* mem_vals = (const float*)d_in[4];  // [B,M,C,RES,RES]
  const int*   mem_valid = (const int*)d_in[5];   // [B,M] bool->int
  const int*   mem_count = (const int*)d_in[6];   // [B]
  float* out = (float*)d_out;

  hipLaunchKernelGGL(hlmm_keys_kernel, dim3(Bn * Mn), dim3(Cn / 4), 0, stream,
                     src_keys, mem_keys, src_valid, mem_valid, mem_count, out);
  hipLaunchKernelGGL(hlmm_values_kernel, dim3(Bn * Mn, VAL_CHUNKS), dim3(VTHREADS),
                     0, stream, src_vals, src_valid, mem_vals, mem_count, out);
}